// MessageLayer_58308476010696
// MI455X (gfx1250) — compile-verified
//
#include <hip/hip_runtime.h>

typedef __attribute__((ext_vector_type(2))) float v2f;
typedef __attribute__((ext_vector_type(8))) float v8f;

#define N_NODES  8192
#define N_EDGES  131072
#define EDGE_DIM 16
#define HIDDEN   32
#define MESSAGE  64
// P columns: 16*64 weight-derived + 64 bias-derived
#define PCOLS    1088
#define NTILES_N 68      // 1088 / 16
#define NTILES_M 512     // 8192 / 16

// ---------------------------------------------------------------------------
// Stage 0: pack W_edge [16, 64*32] + b_edge [64*32] into Wt [32, 1088] where
//   Wt[h, d*64+m]   = W_edge[d, m*32+h]
//   Wt[h, 1024+m]   = b_edge[m*32+h]
// so that stage 1 is a plain row-major GEMM  P = hidden @ Wt.
// ---------------------------------------------------------------------------
__global__ void pack_wt_kernel(const float* __restrict__ W,
                               const float* __restrict__ b,
                               float* __restrict__ Wt) {
    int i = blockIdx.x * blockDim.x + threadIdx.x;
    if (i >= HIDDEN * PCOLS) return;
    int h = i / PCOLS;
    int c = i % PCOLS;
    float v;
    if (c < EDGE_DIM * MESSAGE) {
        int d = c >> 6;          // c / 64
        int m = c & 63;          // c % 64
        v = W[d * (MESSAGE * HIDDEN) + m * HIDDEN + h];
    } else {
        int m = c - EDGE_DIM * MESSAGE;
        v = b[m * HIDDEN + h];
    }
    Wt[i] = v;
}

// ---------------------------------------------------------------------------
// Stage 1: P[8192, 1088] = hidden[8192, 32] @ Wt[32, 1088], full f32 via
// V_WMMA_F32_16X16X4_F32. One wave per 16x16 output tile, 8 WMMA steps (K=32).
//
// A 16x4 f32 layout (ISA 7.12.2): lanes 0-15 -> M=lane, {K=k0, k0+1};
//                                 lanes 16-31 -> M=lane-16, {K=k0+2, k0+3}.
// B 4x16: rows striped across lanes within a VGPR (mirrors A's K split).
// C/D 16x16 f32: VGPR i -> M = i + 8*(lane>=16), N = lane&15.
// ---------------------------------------------------------------------------
__global__ __launch_bounds__(128)
void node_precompute_wmma(const float* __restrict__ hidden,
                          const float* __restrict__ Wt,
                          float* __restrict__ P) {
    const int lane = threadIdx.x & 31;
    const int wave = threadIdx.x >> 5;
    const int tile = blockIdx.x * 4 + wave;          // 34816 tiles total
    const int tn = tile % NTILES_N;
    const int tm = tile / NTILES_N;

    const int r16   = lane & 15;                     // M (for A) / N (for B,C)
    const int khalf = (lane >> 4) * 2;               // 0 or 2
    const int node  = tm * 16 + r16;
    const int col   = tn * 16 + r16;

    v8f c = {};
#pragma unroll
    for (int k0 = 0; k0 < HIDDEN; k0 += 4) {
        // A fragment: two consecutive K values for this lane's row (8B aligned)
        v2f a = *(const v2f*)(hidden + node * HIDDEN + k0 + khalf);
        // B fragment: two K rows of Wt at this lane's column
        v2f bf;
        bf.x = Wt[(k0 + khalf) * PCOLS + col];
        bf.y = Wt[(k0 + khalf + 1) * PCOLS + col];
        c = __builtin_amdgcn_wmma_f32_16x16x4_f32(
                /*neg_a=*/false, a, /*neg_b=*/false, bf,
                /*c_mod=*/(short)0, c, /*reuse_a=*/false, /*reuse_b=*/false);
    }

    const int mrow0 = tm * 16 + (lane >> 4) * 8;
    const int cbase = tn * 16 + r16;
#pragma unroll
    for (int i = 0; i < 8; ++i)
        P[(mrow0 + i) * PCOLS + cbase] = c[i];
}

// ---------------------------------------------------------------------------
// Zero the output (harness poisons d_out; atomics need a clean accumulator).
// ---------------------------------------------------------------------------
__global__ void zero_out_kernel(float4* __restrict__ out) {
    int i = blockIdx.x * blockDim.x + threadIdx.x;   // 131072 float4s
    out[i] = make_float4(0.f, 0.f, 0.f, 0.f);
}

// ---------------------------------------------------------------------------
// Stage 2: one wave per edge. msgs[e] = ef[e] (1x16) @ P[src(e)] (16x64) + bias
// row, then atomic scatter-add into out[dst(e)]. Lane l owns m = l and l+32.
// P-row reads are coalesced 128B lines, served from L2 (P is L2-resident).
// ---------------------------------------------------------------------------
__global__ __launch_bounds__(256)
void edge_message_scatter(const float* __restrict__ ef,
                          const int*   __restrict__ esrc,
                          const int*   __restrict__ edst,
                          const float* __restrict__ P,
                          float*       __restrict__ out) {
    const int lane = threadIdx.x & 31;
    const int wave = threadIdx.x >> 5;
    const int e = blockIdx.x * 8 + wave;             // N_EDGES = 16384*8 exactly

    const int src = esrc[e];
    const int dst = edst[e];
    const float* __restrict__ prow = P + src * PCOLS;

    // Load the 16 edge features once across the wave, broadcast via shuffles.
    float wl = (lane < EDGE_DIM) ? ef[e * EDGE_DIM + lane] : 0.f;

    // Bias-derived columns of P seed the accumulators.
    float acc0 = prow[EDGE_DIM * MESSAGE + lane];
    float acc1 = prow[EDGE_DIM * MESSAGE + 32 + lane];

#pragma unroll
    for (int d = 0; d < EDGE_DIM; ++d) {
        const float w = __shfl(wl, d, 32);
        acc0 = fmaf(w, prow[d * MESSAGE + lane],      acc0);
        acc1 = fmaf(w, prow[d * MESSAGE + 32 + lane], acc1);
    }

    atomicAdd(&out[dst * MESSAGE + lane],      acc0);
    atomicAdd(&out[dst * MESSAGE + 32 + lane], acc1);
}

// ---------------------------------------------------------------------------
extern "C" void kernel_launch(void* const* d_in, const int* in_sizes, int n_in,
                              void* d_out, int out_size, void* d_ws, size_t ws_size,
                              hipStream_t stream) {
    const float* edge_features = (const float*)d_in[0];   // [E, 16]
    const float* hidden        = (const float*)d_in[1];   // [N, 32]
    const int*   edge_src      = (const int*)  d_in[2];   // [E]
    const int*   edge_dst      = (const int*)  d_in[3];   // [E]
    const float* W_edge        = (const float*)d_in[4];   // [16, 2048]
    const float* b_edge        = (const float*)d_in[5];   // [2048]
    float* out = (float*)d_out;                           // [N, 64]

    // Workspace layout: P [8192*1088] then Wt [32*1088]  (~35.8 MB total).
    float* P  = (float*)d_ws;
    float* Wt = P + (size_t)N_NODES * PCOLS;

    pack_wt_kernel<<<(HIDDEN * PCOLS + 255) / 256, 256, 0, stream>>>(W_edge, b_edge, Wt);

    node_precompute_wmma<<<(NTILES_M * NTILES_N) / 4, 128, 0, stream>>>(hidden, Wt, P);

    zero_out_kernel<<<(N_NODES * MESSAGE / 4) / 256, 256, 0, stream>>>((float4*)out);

    edge_message_scatter<<<N_EDGES / 8, 256, 0, stream>>>(
        edge_features, edge_src, edge_dst, P, out);
}